// MPN_Atom_70239895159058
// MI455X (gfx1250) — compile-verified
//
#include <hip/hip_runtime.h>
#include <hip/hip_bf16.h>

#define NAtoms 300001
#define NBonds 600001
#define MAXNB  6
#define AF     133
#define BF     147
#define H      150
#define NPAD   160   // padded N (10 tiles of 16)

// K padded to multiple of 4. LDS row stride: multiple of 4 with (LDA/4) odd ->
// conflict-free ds_load_b64 across all 32 lanes (half0 banks {4u,4u+1}, half1 {4u+2,4u+3}).
#define KP_BF  148
#define LDA_BF 148
#define KP_H   152
#define LDA_H  156
#define KP_AH  284   // AF+H = 283 -> 284
#define LDA_AH 284

typedef float v2f __attribute__((ext_vector_type(2)));
typedef float v8f __attribute__((ext_vector_type(8)));

// ---- weight packing: Wp[(k/2)*2*NPAD + 2*n + (k&1)] = zero-padded W[k][n] ----
// Puts the K-pair a lane needs adjacent -> one global_load_b64 per WMMA step.
__global__ void k_pack(const float* __restrict__ W, float* __restrict__ Wp,
                       int Kact, int KP) {
  int idx = blockIdx.x * 256 + threadIdx.x;
  int total = (KP >> 1) * (2 * NPAD);
  if (idx >= total) return;
  int p = idx / (2 * NPAD);
  int rem = idx - p * (2 * NPAD);
  int n = rem >> 1, j = rem & 1;
  int k = 2 * p + j;
  Wp[idx] = (k < Kact && n < H) ? W[k * H + n] : 0.f;
}

// Two stacked 16x16 M-tiles sharing each B fragment.
// A0/A1: per-lane LDS row pointers (pre-offset by m*lda + 2*half).
// Bp: per-lane packed-W pointer (pre-offset by 2*n + 2*NPAD*half).
__device__ __forceinline__ void gemm2(const float* A0, const float* A1,
                                      const float* __restrict__ Bp, int KP,
                                      v8f& c0, v8f& c1) {
#pragma unroll 2
  for (int k0 = 0; k0 < KP; k0 += 4) {
    v2f a0 = *(const v2f*)(A0 + k0);
    v2f a1 = *(const v2f*)(A1 + k0);
    v2f b  = *(const v2f*)(Bp + (k0 >> 1) * (2 * NPAD));
    c0 = __builtin_amdgcn_wmma_f32_16x16x4_f32(false, a0, false, b, (short)0, c0, false, false);
    c1 = __builtin_amdgcn_wmma_f32_16x16x4_f32(false, a1, false, b, (short)0, c1, false, false);
  }
}

// ---- Kernel 1: inp = f_bonds @ W_i ; msg = relu(inp) ------------------------
__global__ void __launch_bounds__(320)
k_gemm_wi(const float* __restrict__ f_bonds, const float* __restrict__ Wp,
          float* __restrict__ inp, float* __restrict__ msg) {
  const int m0 = blockIdx.x * 32;
  __shared__ __align__(16) float Asm[32 * LDA_BF];
  for (int idx = threadIdx.x; idx < 32 * KP_BF; idx += 320) {
    int r = idx / KP_BF, k = idx - r * KP_BF;
    int b = m0 + r;
    Asm[r * LDA_BF + k] = (b < NBonds && k < BF) ? f_bonds[b * BF + k] : 0.f;
  }
  __syncthreads();
  const int lane = threadIdx.x & 31;
  const int half = lane >> 4, m = lane & 15;
  const int n = (threadIdx.x >> 5) * 16 + m;
  const float* A0 = Asm + m * LDA_BF + 2 * half;
  const float* A1 = A0 + 16 * LDA_BF;
  const float* Bp = Wp + 2 * n + 2 * NPAD * half;
  v8f c0 = (v8f)0.f, c1 = (v8f)0.f;
  gemm2(A0, A1, Bp, KP_BF, c0, c1);
  if (n < H) {
#pragma unroll
    for (int v = 0; v < 8; ++v) {
      int b = m0 + v + 8 * half;
      if (b < NBonds) {
        float x = c0[v];
        inp[b * H + n] = x;
        msg[b * H + n] = fmaxf(x, 0.f);
      }
      int b2 = b + 16;
      if (b2 < NBonds) {
        float x = c1[v];
        inp[b2 * H + n] = x;
        msg[b2 * H + n] = fmaxf(x, 0.f);
      }
    }
  }
}

// ---- Kernel 2: a_msg[a] = sum_j msg[a2b[a][j]] ------------------------------
__global__ void __launch_bounds__(160)
k_aggregate(const float* __restrict__ msg, const int* __restrict__ a2b,
            float* __restrict__ a_msg) {
  const int a = blockIdx.x;
  __shared__ int sidx[MAXNB];
  if (threadIdx.x < MAXNB) sidx[threadIdx.x] = a2b[a * MAXNB + threadIdx.x];
  __syncthreads();
  const int h = threadIdx.x;
  if (h < H) {
    float s = 0.f;
#pragma unroll
    for (int j = 0; j < MAXNB; ++j) s += msg[sidx[j] * H + h];
    a_msg[a * H + h] = s;
  }
}

// ---- Kernel 3: msg_out = relu(inp + (a_msg[b2a]-msg[b2revb]) @ W_h) ---------
__global__ void __launch_bounds__(320)
k_msg_update(const float* __restrict__ a_msg, const float* __restrict__ msg_prev,
             const int* __restrict__ b2a, const int* __restrict__ b2revb,
             const float* __restrict__ Wp, const float* __restrict__ inp,
             float* __restrict__ msg_out) {
  const int m0 = blockIdx.x * 32;
  __shared__ __align__(16) float Asm[32 * LDA_H];
  __shared__ int s_ia[32], s_ir[32];
  if (threadIdx.x < 32) {
    int b = m0 + threadIdx.x;
    bool ok = (b < NBonds);
    s_ia[threadIdx.x] = ok ? b2a[b] : 0;
    s_ir[threadIdx.x] = ok ? b2revb[b] : 0;
  }
  __syncthreads();
  for (int idx = threadIdx.x; idx < 32 * KP_H; idx += 320) {
    int r = idx / KP_H, k = idx - r * KP_H;
    int b = m0 + r;
    float v = 0.f;
    if (b < NBonds && k < H)
      v = a_msg[s_ia[r] * H + k] - msg_prev[s_ir[r] * H + k];
    Asm[r * LDA_H + k] = v;
  }
  __syncthreads();
  const int lane = threadIdx.x & 31;
  const int half = lane >> 4, m = lane & 15;
  const int n = (threadIdx.x >> 5) * 16 + m;
  const float* A0 = Asm + m * LDA_H + 2 * half;
  const float* A1 = A0 + 16 * LDA_H;
  const float* Bp = Wp + 2 * n + 2 * NPAD * half;
  v8f c0 = (v8f)0.f, c1 = (v8f)0.f;
  gemm2(A0, A1, Bp, KP_H, c0, c1);
  if (n < H) {
#pragma unroll
    for (int v = 0; v < 8; ++v) {
      int b = m0 + v + 8 * half;
      if (b < NBonds)
        msg_out[b * H + n] = fmaxf(inp[b * H + n] + c0[v], 0.f);
      int b2 = b + 16;
      if (b2 < NBonds)
        msg_out[b2 * H + n] = fmaxf(inp[b2 * H + n] + c1[v], 0.f);
    }
  }
}

// ---- Kernel 4: atom_hiddens = relu(concat(f_atoms, a_msg) @ W_o + b_o) ------
__global__ void __launch_bounds__(320)
k_atom_out(const float* __restrict__ f_atoms, const float* __restrict__ a_msg,
           const float* __restrict__ Wp, const float* __restrict__ b_o,
           float* __restrict__ atom_h) {
  const int m0 = blockIdx.x * 32;
  __shared__ __align__(16) float Asm[32 * LDA_AH];
  for (int idx = threadIdx.x; idx < 32 * KP_AH; idx += 320) {
    int r = idx / KP_AH, k = idx - r * KP_AH;
    int a = m0 + r;
    float v = 0.f;
    if (a < NAtoms) {
      if (k < AF) v = f_atoms[a * AF + k];
      else if (k < AF + H) v = a_msg[a * H + (k - AF)];
    }
    Asm[r * LDA_AH + k] = v;
  }
  __syncthreads();
  const int lane = threadIdx.x & 31;
  const int half = lane >> 4, m = lane & 15;
  const int n = (threadIdx.x >> 5) * 16 + m;
  const float* A0 = Asm + m * LDA_AH + 2 * half;
  const float* A1 = A0 + 16 * LDA_AH;
  const float* Bp = Wp + 2 * n + 2 * NPAD * half;
  v8f c0 = (v8f)0.f, c1 = (v8f)0.f;
  gemm2(A0, A1, Bp, KP_AH, c0, c1);
  if (n < H) {
    float bias = b_o[n];
#pragma unroll
    for (int v = 0; v < 8; ++v) {
      int a = m0 + v + 8 * half;
      if (a < NAtoms) atom_h[a * H + n] = fmaxf(c0[v] + bias, 0.f);
      int a2 = a + 16;
      if (a2 < NAtoms) atom_h[a2 * H + n] = fmaxf(c1[v] + bias, 0.f);
    }
  }
}

// ---- Kernel 5: per-molecule attention readout -------------------------------
// e[m] = sum_i sigmoid(bh_i @ W_jl + b_jl) * (bh_i @ W_il + b_il)
// One K-loop: 2 A frags + 2 B frags -> 4 WMMAs.
__global__ void __launch_bounds__(320)
k_attention(const float* __restrict__ atom_h,
            const float* __restrict__ Wpjl, const float* __restrict__ b_jl,
            const float* __restrict__ Wpil, const float* __restrict__ b_il,
            float* __restrict__ out, int apm) {
  const int mol = blockIdx.x;
  __shared__ __align__(16) float Asm[32 * LDA_H];
  __shared__ float pairs[320];
  for (int idx = threadIdx.x; idx < 32 * KP_H; idx += 320) {
    int r = idx / KP_H, k = idx - r * KP_H;
    float v = 0.f;
    if (r < apm && k < H)
      v = atom_h[(1 + mol * apm + r) * H + k];
    Asm[r * LDA_H + k] = v;
  }
  __syncthreads();
  const int lane = threadIdx.x & 31;
  const int half = lane >> 4, m = lane & 15;
  const int n = (threadIdx.x >> 5) * 16 + m;
  const float* A0 = Asm + m * LDA_H + 2 * half;
  const float* A1 = A0 + 16 * LDA_H;
  const float* Bj = Wpjl + 2 * n + 2 * NPAD * half;
  const float* Bi = Wpil + 2 * n + 2 * NPAD * half;
  v8f p0 = (v8f)0.f, p1 = (v8f)0.f, q0 = (v8f)0.f, q1 = (v8f)0.f;
#pragma unroll 2
  for (int k0 = 0; k0 < KP_H; k0 += 4) {
    v2f a0 = *(const v2f*)(A0 + k0);
    v2f a1 = *(const v2f*)(A1 + k0);
    v2f bj = *(const v2f*)(Bj + (k0 >> 1) * (2 * NPAD));
    v2f bi = *(const v2f*)(Bi + (k0 >> 1) * (2 * NPAD));
    p0 = __builtin_amdgcn_wmma_f32_16x16x4_f32(false, a0, false, bj, (short)0, p0, false, false);
    p1 = __builtin_amdgcn_wmma_f32_16x16x4_f32(false, a1, false, bj, (short)0, p1, false, false);
    q0 = __builtin_amdgcn_wmma_f32_16x16x4_f32(false, a0, false, bi, (short)0, q0, false, false);
    q1 = __builtin_amdgcn_wmma_f32_16x16x4_f32(false, a1, false, bi, (short)0, q1, false, false);
  }
  float partial = 0.f;
  if (n < H) {
    float bj = b_jl[n], bi = b_il[n];
#pragma unroll
    for (int v = 0; v < 8; ++v) {
      int i0 = v + 8 * half;
      if (i0 < apm) {
        float pp = p0[v] + bj, qq = q0[v] + bi;
        partial += qq / (1.f + __expf(-pp));
      }
      int i1 = i0 + 16;
      if (i1 < apm) {
        float pp = p1[v] + bj, qq = q1[v] + bi;
        partial += qq / (1.f + __expf(-pp));
      }
    }
  }
  pairs[threadIdx.x] = partial;
  __syncthreads();
  if (threadIdx.x < H) {
    int nn = threadIdx.x, w = nn >> 4, c = nn & 15;
    out[mol * H + nn] = pairs[w * 32 + c] + pairs[w * 32 + 16 + c];
  }
}

extern "C" void kernel_launch(void* const* d_in, const int* in_sizes, int n_in,
                              void* d_out, int out_size, void* d_ws, size_t ws_size,
                              hipStream_t stream) {
  (void)in_sizes; (void)n_in; (void)out_size; (void)ws_size;
  const float* f_atoms = (const float*)d_in[0];
  const float* f_bonds = (const float*)d_in[1];
  const int*   a2b     = (const int*)d_in[2];
  const int*   b2a     = (const int*)d_in[3];
  const int*   b2revb  = (const int*)d_in[4];
  const float* W_i     = (const float*)d_in[5];
  const float* W_h     = (const float*)d_in[6];
  const float* W_o     = (const float*)d_in[7];
  const float* b_o     = (const float*)d_in[8];
  const float* W_il    = (const float*)d_in[9];
  const float* b_il    = (const float*)d_in[10];
  const float* W_jl    = (const float*)d_in[11];
  const float* b_jl    = (const float*)d_in[12];
  const int n_mols = 12000;
  const int apm    = 25;

  const size_t NBH = (size_t)NBonds * H;
  const size_t NAH = (size_t)NAtoms * H;
  const size_t SZ_WI = (size_t)(KP_BF / 2) * 2 * NPAD;   // 23680
  const size_t SZ_WH = (size_t)(KP_H  / 2) * 2 * NPAD;   // 24320
  const size_t SZ_WO = (size_t)(KP_AH / 2) * 2 * NPAD;   // 45440

  float* ws     = (float*)d_ws;
  float* inp    = ws;               // [NB, H]
  float* msg0   = inp  + NBH;       // [NB, H]
  float* msg1   = msg0 + NBH;       // [NB, H]
  float* amsg   = msg1 + NBH;       // [NA, H]
  float* atom_h = amsg + NAH;       // [NA, H]
  float* wpi    = atom_h + NAH;
  float* wph    = wpi  + SZ_WI;
  float* wpo    = wph  + SZ_WH;
  float* wpjl   = wpo  + SZ_WO;
  float* wpil   = wpjl + SZ_WH;

  // pack weights (zero-padded, K-pair interleaved)
  k_pack<<<(int)((SZ_WI + 255) / 256), 256, 0, stream>>>(W_i,  wpi,  BF,     KP_BF);
  k_pack<<<(int)((SZ_WH + 255) / 256), 256, 0, stream>>>(W_h,  wph,  H,      KP_H);
  k_pack<<<(int)((SZ_WO + 255) / 256), 256, 0, stream>>>(W_o,  wpo,  AF + H, KP_AH);
  k_pack<<<(int)((SZ_WH + 255) / 256), 256, 0, stream>>>(W_jl, wpjl, H,      KP_H);
  k_pack<<<(int)((SZ_WH + 255) / 256), 256, 0, stream>>>(W_il, wpil, H,      KP_H);

  const int NBT = (NBonds + 31) / 32;   // 18751
  const int NAT = (NAtoms + 31) / 32;   // 9376

  k_gemm_wi<<<NBT, 320, 0, stream>>>(f_bonds, wpi, inp, msg0);

  // depth iteration 1
  k_aggregate<<<NAtoms, 160, 0, stream>>>(msg0, a2b, amsg);
  k_msg_update<<<NBT, 320, 0, stream>>>(amsg, msg0, b2a, b2revb, wph, inp, msg1);
  // depth iteration 2
  k_aggregate<<<NAtoms, 160, 0, stream>>>(msg1, a2b, amsg);
  k_msg_update<<<NBT, 320, 0, stream>>>(amsg, msg1, b2a, b2revb, wph, inp, msg0);
  // final aggregation + readout
  k_aggregate<<<NAtoms, 160, 0, stream>>>(msg0, a2b, amsg);
  k_atom_out<<<NAT, 320, 0, stream>>>(f_atoms, amsg, wpo, b_o, atom_h);
  k_attention<<<n_mols, 320, 0, stream>>>(atom_h, wpjl, b_jl, wpil, b_il,
                                          (float*)d_out, apm);
}